// DGCNN_12421045420215
// MI455X (gfx1250) — compile-verified
//
#include <hip/hip_runtime.h>
#include <hip/hip_bf16.h>
#include <math.h>

// ---------------------------------------------------------------------------
// DGCNN forward for MI455X (gfx1250, wave32).
//  - GCN layer GEMMs use native f32 WMMA: V_WMMA_F32_16X16X4_F32 (exact f32).
//  - Edge scatter is the bandwidth-bound phase: wave = one edge, lane = feature
//    (HIDDEN=32 == wave32), fully coalesced gather + global f32 atomics.
//  - Sort-pool + conv/fc head: one workgroup per graph, LDS staged.
// ---------------------------------------------------------------------------

typedef float v2f __attribute__((ext_vector_type(2)));
typedef float v8f __attribute__((ext_vector_type(8)));

#define IN_FEAT 128
#define HIDDEN  32
#define NPG     100   // nodes per graph
#define KTOP    35

// ---------------------------------------------------------------------------
// h = A[M x K] (row stride lda) @ W[K x 32], f32 WMMA 16x16x4.
// One wave -> 16 rows x 32 cols (two 16x16 accumulators). Block = 8 waves.
// Requires M % 16 == 0 (true: 100000).
// ---------------------------------------------------------------------------
__global__ void gcn_gemm_f32_wmma(const float* __restrict__ A, int lda,
                                  const float* __restrict__ W, int K,
                                  float* __restrict__ H, int M) {
  __shared__ float sW[IN_FEAT * HIDDEN];   // up to 128x32 = 16KB

  const int tid = threadIdx.x;
  for (int i = tid; i < K * HIDDEN; i += blockDim.x) sW[i] = W[i];
  __syncthreads();

  const int wave = tid >> 5;
  const int lane = tid & 31;
  const int m = lane & 15;      // row within tile
  const int g = lane >> 4;      // lane group (K striping)
  const int row0 = (blockIdx.x * 8 + wave) * 16;

  if (row0 < M) {               // wave-uniform: EXEC stays all-ones for WMMA
    v8f c0 = {};
    v8f c1 = {};
    const float* __restrict__ arow = A + (size_t)(row0 + m) * lda;

    for (int kb = 0; kb < K; kb += 4) {
      const int ka = kb + 2 * g;                 // A/B K index for this lane grp
      v2f a = *(const v2f*)(arow + ka);          // A: K=ka, ka+1 (8B aligned)
      v2f b0, b1;
      b0.x = sW[(ka + 0) * HIDDEN + m];          // B cols 0..15
      b0.y = sW[(ka + 1) * HIDDEN + m];
      b1.x = sW[(ka + 0) * HIDDEN + 16 + m];     // B cols 16..31
      b1.y = sW[(ka + 1) * HIDDEN + 16 + m];
      c0 = __builtin_amdgcn_wmma_f32_16x16x4_f32(false, a, false, b0,
                                                 (short)0, c0, false, false);
      c1 = __builtin_amdgcn_wmma_f32_16x16x4_f32(false, a, false, b1,
                                                 (short)0, c1, false, false);
    }

    // C/D layout: VGPR v, lane-group g -> row = row0 + v + 8g, col = m (+16)
    for (int v = 0; v < 8; ++v) {
      const int r = row0 + v + 8 * g;
      H[(size_t)r * HIDDEN + m]      = c0[v];
      H[(size_t)r * HIDDEN + 16 + m] = c1[v];
    }
  }
}

// ---------------------------------------------------------------------------
// Degree (incl. self loop) and inverse-sqrt normalization.
// ---------------------------------------------------------------------------
__global__ void init_deg(float* __restrict__ deg, int N) {
  int i = blockIdx.x * blockDim.x + threadIdx.x;
  if (i < N) deg[i] = 1.0f;                      // self-loop
}

__global__ void count_deg(const int* __restrict__ dst, float* __restrict__ deg,
                          long long E) {
  long long e = (long long)blockIdx.x * blockDim.x + threadIdx.x;
  if (e < E) atomicAdd(&deg[dst[e]], 1.0f);
}

__global__ void make_dis(const float* __restrict__ deg, float* __restrict__ dis,
                         int N) {
  int i = blockIdx.x * blockDim.x + threadIdx.x;
  if (i < N) dis[i] = rsqrtf(deg[i]);
}

// ---------------------------------------------------------------------------
// agg[dst] += dis[src]*dis[dst] * h[src].  wave = 1 edge, lane = feature.
// ---------------------------------------------------------------------------
__global__ void edge_scatter(const int* __restrict__ src,
                             const int* __restrict__ dst,
                             const float* __restrict__ dis,
                             const float* __restrict__ h,
                             float* __restrict__ agg, long long E) {
  long long idx = (long long)blockIdx.x * blockDim.x + threadIdx.x;
  long long e = idx >> 5;
  int f = (int)(idx & 31);
  if (e >= E) return;
  int s = src[e];
  int d = dst[e];
  float coef = dis[s] * dis[d];
  atomicAdd(&agg[(size_t)d * HIDDEN + f], coef * h[(size_t)s * HIDDEN + f]);
}

// ---------------------------------------------------------------------------
// feat[:, 32*layer + f] = tanh(agg + h/deg + bias)
// ---------------------------------------------------------------------------
__global__ void combine_act(const float* __restrict__ agg,
                            const float* __restrict__ h,
                            const float* __restrict__ deg,
                            const float* __restrict__ bias,
                            float* __restrict__ feat, int N, int layer) {
  long long idx = (long long)blockIdx.x * blockDim.x + threadIdx.x;
  if (idx >= (long long)N * HIDDEN) return;
  int n = (int)(idx >> 5);
  int f = (int)(idx & 31);
  float val = agg[idx] + h[idx] / deg[n] + bias[f];
  feat[(size_t)n * IN_FEAT + layer * HIDDEN + f] = tanhf(val);
}

// ---------------------------------------------------------------------------
// Per-graph: stable top-K sort by last channel, Conv1d(128->16,k1) + ReLU,
// MaxPool(2), Conv1d(16->32,k5) + ReLU, fc1(416->128)+ReLU, fc2(128->1),
// sigmoid.  One workgroup of 128 threads per graph.
// ---------------------------------------------------------------------------
__global__ void sortpool_head(const float* __restrict__ feat,
                              const float* __restrict__ c1w,
                              const float* __restrict__ c1b,
                              const float* __restrict__ c2w,
                              const float* __restrict__ c2b,
                              const float* __restrict__ f1w,
                              const float* __restrict__ f1b,
                              const float* __restrict__ f2w,
                              const float* __restrict__ f2b,
                              float* __restrict__ out) {
  __shared__ float sLast[NPG];
  __shared__ int   sIdx[KTOP];
  __shared__ float sF[KTOP][IN_FEAT];
  __shared__ float y1[16][KTOP];
  __shared__ float y2[16][17];
  __shared__ float y3[32 * 13];
  __shared__ float z[128];

  const int g = blockIdx.x;
  const int tid = threadIdx.x;
  const float* __restrict__ fg = feat + (size_t)g * NPG * IN_FEAT;

  if (tid < NPG) sLast[tid] = fg[tid * IN_FEAT + (IN_FEAT - 1)];
  __syncthreads();

  // rank == position in stable descending argsort (ties -> lower index first)
  if (tid < NPG) {
    float v = sLast[tid];
    int r = 0;
    for (int j = 0; j < NPG; ++j) {
      float u = sLast[j];
      r += (u > v) || (u == v && j < tid);
    }
    if (r < KTOP) sIdx[r] = tid;
  }
  __syncthreads();

  for (int t = tid; t < KTOP * IN_FEAT; t += blockDim.x) {
    int k = t >> 7, c = t & 127;
    sF[k][c] = fg[sIdx[k] * IN_FEAT + c];
  }
  __syncthreads();

  // conv1 (1x1): y1[o][k] = relu(b[o] + sum_c c1w[o,c] * sF[k][c])
  for (int t = tid; t < 16 * KTOP; t += blockDim.x) {
    int o = t / KTOP, k = t % KTOP;
    float acc = c1b[o];
    const float* wr = c1w + o * IN_FEAT;
    for (int c = 0; c < IN_FEAT; ++c) acc = fmaf(wr[c], sF[k][c], acc);
    y1[o][k] = fmaxf(acc, 0.f);
  }
  __syncthreads();

  // maxpool(2): 35 -> 17 (drop trailing)
  for (int t = tid; t < 16 * 17; t += blockDim.x) {
    int o = t / 17, l = t % 17;
    y2[o][l] = fmaxf(y1[o][2 * l], y1[o][2 * l + 1]);
  }
  __syncthreads();

  // conv2 (k=5, valid): 17 -> 13; y3[o*13+p]
  for (int t = tid; t < 32 * 13; t += blockDim.x) {
    int o = t / 13, p = t % 13;
    float acc = c2b[o];
    for (int c = 0; c < 16; ++c) {
      const float* wr = c2w + (o * 16 + c) * 5;
      for (int i = 0; i < 5; ++i) acc = fmaf(wr[i], y2[c][p + i], acc);
    }
    y3[o * 13 + p] = fmaxf(acc, 0.f);
  }
  __syncthreads();

  // fc1: z[j] = relu(b + sum_u f1w[j,u] * y3[u])   (416 inputs)
  if (tid < 128) {
    float acc = f1b[tid];
    const float* wr = f1w + tid * (32 * 13);
    for (int u = 0; u < 32 * 13; ++u) acc = fmaf(wr[u], y3[u], acc);
    z[tid] = fmaxf(acc, 0.f);
  }
  __syncthreads();

  if (tid == 0) {
    float acc = f2b[0];
    for (int j = 0; j < 128; ++j) acc = fmaf(f2w[j], z[j], acc);
    out[g] = 1.f / (1.f + expf(-acc));
  }
}

// ---------------------------------------------------------------------------
extern "C" void kernel_launch(void* const* d_in, const int* in_sizes, int n_in,
                              void* d_out, int out_size, void* d_ws, size_t ws_size,
                              hipStream_t stream) {
  const float* x   = (const float*)d_in[0];
  const int*   ei  = (const int*)d_in[1];       // [2, E]: src then dst
  const float* w[4] = {(const float*)d_in[3], (const float*)d_in[5],
                       (const float*)d_in[7], (const float*)d_in[9]};
  const float* b[4] = {(const float*)d_in[4], (const float*)d_in[6],
                       (const float*)d_in[8], (const float*)d_in[10]};
  const float* c1w = (const float*)d_in[11];
  const float* c1b = (const float*)d_in[12];
  const float* c2w = (const float*)d_in[13];
  const float* c2b = (const float*)d_in[14];
  const float* f1w = (const float*)d_in[15];
  const float* f1b = (const float*)d_in[16];
  const float* f2w = (const float*)d_in[17];
  const float* f2b = (const float*)d_in[18];
  float* out = (float*)d_out;

  const int       N = in_sizes[0] / IN_FEAT;    // 100000
  const long long E = in_sizes[1] / 2;          // 3200000
  const int       G = out_size;                 // 1000

  // workspace carve
  float* feat = (float*)d_ws;                       // [N,128]
  float* h    = feat + (size_t)N * IN_FEAT;         // [N,32]
  float* agg  = h    + (size_t)N * HIDDEN;          // [N,32]
  float* deg  = agg  + (size_t)N * HIDDEN;          // [N]
  float* dis  = deg  + N;                           // [N]

  const int* src = ei;
  const int* dst = ei + E;

  // normalization terms (once)
  init_deg<<<(N + 255) / 256, 256, 0, stream>>>(deg, N);
  count_deg<<<(unsigned)((E + 255) / 256), 256, 0, stream>>>(dst, deg, E);
  make_dis<<<(N + 255) / 256, 256, 0, stream>>>(deg, dis, N);

  const long long nodeFeatThreads = (long long)N * HIDDEN;
  const long long edgeThreads     = E * HIDDEN;

  for (int l = 0; l < 4; ++l) {
    hipMemsetAsync(agg, 0, (size_t)N * HIDDEN * sizeof(float), stream);

    const float* A = (l == 0) ? x : (feat + (size_t)(l - 1) * HIDDEN);
    const int K = (l == 0) ? IN_FEAT : HIDDEN;
    const int tiles = N / 16;                       // 6250, N % 16 == 0
    gcn_gemm_f32_wmma<<<(tiles + 7) / 8, 256, 0, stream>>>(A, IN_FEAT, w[l], K,
                                                           h, N);

    edge_scatter<<<(unsigned)((edgeThreads + 255) / 256), 256, 0, stream>>>(
        src, dst, dis, h, agg, E);

    combine_act<<<(unsigned)((nodeFeatThreads + 255) / 256), 256, 0, stream>>>(
        agg, h, deg, b[l], feat, N, l);
  }

  sortpool_head<<<G, 128, 0, stream>>>(feat, c1w, c1b, c2w, c2b, f1w, f1b,
                                       f2w, f2b, out);
}